// Sim_CNN_63204738728591
// MI455X (gfx1250) — compile-verified
//
#include <hip/hip_runtime.h>

#define LSEQ  512
#define D_EMB 300
#define C_LAB 128
#define BATCH 64
#define JPAD  912   // 3*300 padded up to a multiple of 16 for WMMA tiling

typedef __attribute__((ext_vector_type(2))) float v2f;
typedef __attribute__((ext_vector_type(8))) float v8f;

// workspace layout in floats
#define OFF_G    0                       // 3*512 = 1536
#define OFF_BIAS 1536                    // 128
#define OFF_MT   1792                    // 912*128 = 116736
#define OFF_U    (1792 + JPAD * C_LAB)   // 64*912 = 58368

// ---------------------------------------------------------------------------
// Kernel 1: per-token window weights g_k[t] and the fully folded output bias.
// g_k[t] = sum_{l=max(0,t-k+1)}^{min(t,L-k-1)} w_k[l]
// bias'[c'] = fc_b[c'] + sum_k ( S_k * <fc_w[c',k*C:], conv_b> + b_k * sum(fc_w[c',k*C:]) )
// ---------------------------------------------------------------------------
__global__ __launch_bounds__(512) void k_prep(
    const float* __restrict__ fc3_w, const float* __restrict__ fc3_b,
    const float* __restrict__ fc4_w, const float* __restrict__ fc4_b,
    const float* __restrict__ fc5_w, const float* __restrict__ fc5_b,
    const float* __restrict__ fc_w,  const float* __restrict__ fc_b,
    const float* __restrict__ conv_b,
    float* __restrict__ g, float* __restrict__ bias)
{
    const int t = threadIdx.x;                 // 0..511
    const float* wk[3] = {fc3_w, fc4_w, fc5_w};
    const int    nk[3] = {LSEQ - 3, LSEQ - 4, LSEQ - 5};
    __shared__ float sS[3];

    #pragma unroll
    for (int ki = 0; ki < 3; ++ki) {
        const int k = ki + 3;
        int lo = t - k + 1; if (lo < 0) lo = 0;
        int hi = t;         if (hi > nk[ki] - 1) hi = nk[ki] - 1;
        float s = 0.f;
        for (int l = lo; l <= hi; ++l) s += wk[ki][l];
        g[ki * LSEQ + t] = s;
    }
    if (t < 3) {                               // S_k = sum_l w_k[l]
        float s = 0.f;
        for (int l = 0; l < nk[t]; ++l) s += wk[t][l];
        sS[t] = s;
    }
    __syncthreads();
    if (t < C_LAB) {
        const float bk[3] = {fc3_b[0], fc4_b[0], fc5_b[0]};
        float b = fc_b[t];
        #pragma unroll
        for (int ki = 0; ki < 3; ++ki) {
            const float* fw = fc_w + t * (3 * C_LAB) + ki * C_LAB;
            float dot = 0.f, sw = 0.f;
            for (int c = 0; c < C_LAB; ++c) { dot = fmaf(fw[c], conv_b[c], dot); sw += fw[c]; }
            b += sS[ki] * dot + bk[ki] * sw;
        }
        bias[t] = b;
    }
}

// ---------------------------------------------------------------------------
// Kernel 2: Mt[j][c'] = sum_c fc_w[c', k*C + c] * conv_w[c, d],  j = k*300 + d
// Rows 900..911 zero-padded. One block per j, one thread per c'.
// ---------------------------------------------------------------------------
__global__ __launch_bounds__(128) void k_foldM(
    const float* __restrict__ conv_w, const float* __restrict__ fc_w,
    float* __restrict__ Mt)
{
    const int j  = blockIdx.x;     // 0..911
    const int cp = threadIdx.x;    // c'
    if (j >= 3 * D_EMB) { Mt[j * C_LAB + cp] = 0.f; return; }   // uniform branch
    const int ki = j / D_EMB;
    const int d  = j - ki * D_EMB;
    __shared__ float col[C_LAB];
    col[cp] = conv_w[cp * D_EMB + d];
    __syncthreads();
    const float* fw = fc_w + cp * (3 * C_LAB) + ki * C_LAB;
    float s = 0.f;
    #pragma unroll 8
    for (int c = 0; c < C_LAB; ++c) s = fmaf(fw[c], col[c], s);
    Mt[j * C_LAB + cp] = s;
}

// ---------------------------------------------------------------------------
// Kernel 3: weighted embedding bag. U[b][k*300+d] = sum_t g_k[t]*embed_w[x[b,t],d]
// One block per batch row; indices + weights staged in LDS; gather stream
// prefetched with global_prefetch_b8.
// ---------------------------------------------------------------------------
__global__ __launch_bounds__(128) void k_bag(
    const int* __restrict__ x, const float* __restrict__ embed_w,
    const float* __restrict__ g, float* __restrict__ U)
{
    const int b   = blockIdx.x;
    const int tid = threadIdx.x;
    __shared__ float sg[3 * LSEQ];
    __shared__ int   sx[LSEQ];
    for (int i = tid; i < 3 * LSEQ; i += 128) sg[i] = g[i];
    for (int i = tid; i < LSEQ;     i += 128) sx[i] = x[b * LSEQ + i];
    __syncthreads();

    float acc[3][3] = {};                       // [d-chunk][k]
    for (int t = 0; t < LSEQ; ++t) {
        const int   row = sx[t];
        const float g0 = sg[t], g1 = sg[LSEQ + t], g2 = sg[2 * LSEQ + t];
        const float* e = embed_w + (long)row * D_EMB;
        if (t + 6 < LSEQ) {                     // prefetch a future gathered row
            const float* pn = embed_w + (long)sx[t + 6] * D_EMB;
            __builtin_prefetch(pn + tid * 2, 0, 1);
        }
        #pragma unroll
        for (int cch = 0; cch < 3; ++cch) {
            const int d = tid + cch * 128;
            if (d < D_EMB) {
                const float v = e[d];
                acc[cch][0] = fmaf(g0, v, acc[cch][0]);
                acc[cch][1] = fmaf(g1, v, acc[cch][1]);
                acc[cch][2] = fmaf(g2, v, acc[cch][2]);
            }
        }
    }
    float* ub = U + b * JPAD;
    #pragma unroll
    for (int cch = 0; cch < 3; ++cch) {
        const int d = tid + cch * 128;
        if (d < D_EMB) {
            ub[0 * D_EMB + d] = acc[cch][0];
            ub[1 * D_EMB + d] = acc[cch][1];
            ub[2 * D_EMB + d] = acc[cch][2];
        }
    }
    if (tid < JPAD - 3 * D_EMB) ub[3 * D_EMB + tid] = 0.f;   // zero the K pad
}

// ---------------------------------------------------------------------------
// Kernel 4: out[64,128] = U[64,912] @ Mt[912,128] + bias, fp32 WMMA.
// One wave32 per 16x16 output tile (4 x 8 tiles), K stepped by 4 via
// V_WMMA_F32_16X16X4_F32 with the documented VGPR layouts:
//   A 16x4:  lane<16 -> K={0,1}, lane>=16 -> K={2,3}, row M = lane&15
//   B 4x16:  v0 = row K=(half*2), v1 = row K=(half*2+1), col N = lane&15
//   C 16x16: VGPR r -> M = r + 8*half, N = lane&15
// ---------------------------------------------------------------------------
__global__ __launch_bounds__(32) void k_gemm(
    const float* __restrict__ U, const float* __restrict__ Mt,
    const float* __restrict__ bias, float* __restrict__ out)
{
    const int tile = blockIdx.x;        // 0..31
    const int m0   = (tile & 3) * 16;   // 64 rows  -> 4 tiles
    const int n0   = (tile >> 2) * 16;  // 128 cols -> 8 tiles
    const int lane = threadIdx.x;
    const int half = lane >> 4;
    const int lr   = lane & 15;

    v8f acc = {};
    const float* arow = U  + (m0 + lr) * JPAD + half * 2;          // A fragment base
    const float* bcol = Mt + (n0 + lr) + (half * 2) * C_LAB;       // B fragment base

    #pragma unroll 4
    for (int k0 = 0; k0 < JPAD; k0 += 4) {
        v2f a, bb;
        a.x  = arow[k0];
        a.y  = arow[k0 + 1];
        bb.x = bcol[k0 * C_LAB];
        bb.y = bcol[k0 * C_LAB + C_LAB];
        acc = __builtin_amdgcn_wmma_f32_16x16x4_f32(
            /*neg_a=*/false, a, /*neg_b=*/false, bb,
            /*c_mod=*/(short)0, acc, /*reuse_a=*/false, /*reuse_b=*/false);
    }

    const float bn = bias[n0 + lr];
    #pragma unroll
    for (int r = 0; r < 8; ++r) {
        const int m = m0 + r + half * 8;
        out[m * C_LAB + n0 + lr] = acc[r] + bn;
    }
}

// ---------------------------------------------------------------------------
extern "C" void kernel_launch(void* const* d_in, const int* in_sizes, int n_in,
                              void* d_out, int out_size, void* d_ws, size_t ws_size,
                              hipStream_t stream) {
    (void)in_sizes; (void)n_in; (void)out_size; (void)ws_size;
    const int*   x       = (const int*)  d_in[0];
    const float* embed_w = (const float*)d_in[1];
    const float* conv_w  = (const float*)d_in[2];
    const float* conv_b  = (const float*)d_in[3];
    const float* fc3_w   = (const float*)d_in[4];
    const float* fc3_b   = (const float*)d_in[5];
    const float* fc4_w   = (const float*)d_in[6];
    const float* fc4_b   = (const float*)d_in[7];
    const float* fc5_w   = (const float*)d_in[8];
    const float* fc5_b   = (const float*)d_in[9];
    const float* fc_w    = (const float*)d_in[10];
    const float* fc_b    = (const float*)d_in[11];
    float* out = (float*)d_out;
    float* ws  = (float*)d_ws;

    float* g    = ws + OFF_G;
    float* bias = ws + OFF_BIAS;
    float* Mt   = ws + OFF_MT;
    float* U    = ws + OFF_U;

    hipLaunchKernelGGL(k_prep, dim3(1), dim3(512), 0, stream,
                       fc3_w, fc3_b, fc4_w, fc4_b, fc5_w, fc5_b,
                       fc_w, fc_b, conv_b, g, bias);
    hipLaunchKernelGGL(k_foldM, dim3(JPAD), dim3(128), 0, stream,
                       conv_w, fc_w, Mt);
    hipLaunchKernelGGL(k_bag, dim3(BATCH), dim3(128), 0, stream,
                       x, embed_w, g, U);
    hipLaunchKernelGGL(k_gemm, dim3(32), dim3(32), 0, stream,
                       U, Mt, bias, out);
}